// ts_std_65455301591102
// MI455X (gfx1250) — compile-verified
//
#include <hip/hip_runtime.h>
#include <stdint.h>

// Problem constants (static in the reference's setup_inputs()).
#define BDIM 64
#define TDIM 2048
#define FDIM 256
#define WIN  32
#define SDIM (TDIM - WIN + 1)      // 2017 output windows (stride 1)
#define CHUNK 253                  // outputs per time-chunk
#define NC 8                       // chunks per batch (8*253 = 2024 >= 2017)
#define TILE_T 32                  // time rows per LDS tile (== WIN)
#define TILE_ELEMS (TILE_T * FDIM) // 8192 floats
#define TILE_BYTES (TILE_ELEMS * 4)// 32 KB per buffer, 64 KB double-buffered
#define V16_PER_THREAD ((TILE_BYTES / 16) / 256) // 8 x b128 per thread per tile

__global__ __launch_bounds__(256) void ts_std_async_kernel(
    const float* __restrict__ x, float* __restrict__ out) {
  __shared__ float smem[2 * TILE_ELEMS];

  const int tid   = threadIdx.x;          // feature index
  const int chunk = blockIdx.x;           // time chunk
  const int b     = blockIdx.y;           // batch

  const int OS = chunk * CHUNK;           // first output window of this chunk
  int n_out = SDIM - OS; if (n_out > CHUNK) n_out = CHUNK;
  const int n_in = n_out + (WIN - 1);
  const int nt   = (n_in + TILE_T - 1) / TILE_T;

  // Uniform 64-bit base for this batch; per-lane 32-bit offsets (max 2 MB).
  const char* gbase = (const char*)x + (size_t)b * TDIM * FDIM * 4;
  const uint32_t lds0 = (uint32_t)(uintptr_t)&smem[0];

  // Issue one 32x256 f32 tile as async global->LDS b128 copies.
  auto issue = [&](int k) {
    const int t0 = OS + k * TILE_T;
    const uint32_t lbase = lds0 + (uint32_t)(k & 1) * TILE_BYTES;
#pragma unroll
    for (int q = 0; q < V16_PER_THREAD; ++q) {
      int e   = tid + q * 256;   // 16-byte packet index within tile [0,2048)
      int row = e >> 6;          // 64 packets per 1 KB time-row
      int col = e & 63;
      int t   = t0 + row;
      t = (t < TDIM - 1) ? t : (TDIM - 1);  // clamp tail (never stored)
      uint32_t goff = (uint32_t)t * (FDIM * 4) + (uint32_t)col * 16u;
      uint32_t loff = lbase + (uint32_t)e * 16u;
      asm volatile("global_load_async_to_lds_b128 %0, %1, %2"
                   :: "v"(loff), "v"(goff), "s"(gbase) : "memory");
    }
  };

  issue(0);

  // 32-deep register circular window history (static indices after unroll).
  float h[WIN];
#pragma unroll
  for (int j = 0; j < WIN; ++j) h[j] = 0.0f;
  float s1 = 0.0f, s2 = 0.0f;
  const float inv_w = 1.0f / (float)WIN;
  float* outp = out + ((size_t)b * SDIM + OS) * FDIM + tid;

  for (int k = 0; k < nt; ++k) {
    if (k + 1 < nt) {
      issue(k + 1);                                   // prefetch next tile
      asm volatile("s_wait_asynccnt 0x8" ::: "memory"); // tile k landed (in-order)
    } else {
      asm volatile("s_wait_asynccnt 0x0" ::: "memory");
    }
    __syncthreads();  // all waves' tile-k copies visible

    const float* buf = &smem[(k & 1) * TILE_ELEMS];
#pragma unroll
    for (int j = 0; j < WIN; ++j) {
      float xn = buf[j * FDIM + tid];
      s1 += xn - h[j];
      s2 += xn * xn - h[j] * h[j];
      h[j] = xn;
      int srel = k * TILE_T + j - (WIN - 1);
      if (srel >= 0 && srel < n_out) {
        float mean = s1 * inv_w;
        float var  = fmaf(-mean, mean, s2 * inv_w);
        var = var > 0.0f ? var : 0.0f;               // also kills NaN -> 0 path
        outp[(size_t)srel * FDIM] = __builtin_sqrtf(var);
      }
    }
    __syncthreads();  // buffer (k&1) fully consumed before it is re-targeted
  }
}

extern "C" void kernel_launch(void* const* d_in, const int* in_sizes, int n_in_args,
                              void* d_out, int out_size, void* d_ws, size_t ws_size,
                              hipStream_t stream) {
  (void)in_sizes; (void)n_in_args; (void)out_size; (void)d_ws; (void)ws_size;
  const float* x = (const float*)d_in[0];   // [64, 2048, 256] f32
  float* out = (float*)d_out;               // [64, 2017, 256] f32
  dim3 grid(NC, BDIM);
  ts_std_async_kernel<<<grid, 256, 0, stream>>>(x, out);
}